// ScaledPredictionLayer_70385924047344
// MI455X (gfx1250) — compile-verified
//
#include <hip/hip_runtime.h>
#include <hip/hip_bf16.h>

#define TPB 256

// One row per thread. pred row = 28 f32 = 112 B ; scaled row = 6 f32 = 24 B.
// Per-block staging: pred tile = 256*112 = 28672 B (1792 x b128),
//                    scaled tile = 256*24 = 6144 B (384 x b128).
// Inputs are streamed once (~294 MB > 192 MB L2) -> non-temporal hints.

__global__ __launch_bounds__(TPB) void ScaledPredictionLayer_kernel(
    const float* __restrict__ scaled,   // [B,6]
    const float* __restrict__ pred,     // [B,28]
    const float* __restrict__ pmean,    // [6]
    const float* __restrict__ pscale,   // [6]
    const float* __restrict__ tmean,    // [1]
    const float* __restrict__ tscale,   // [1]
    float* __restrict__ out,            // [B]
    int B)
{
    __shared__ float lds_pred[TPB * 28];   // 28 KB
    __shared__ float lds_scaled[TPB * 6];  //  6 KB

    const int tid = threadIdx.x;
    const long long row0 = (long long)blockIdx.x * TPB;
    const bool full_tile = (row0 + TPB) <= (long long)B;

    if (full_tile) {
        // Low 32 bits of the generic pointer == workgroup-relative LDS address.
        const unsigned lp = (unsigned)(size_t)lds_pred;
        const unsigned ls = (unsigned)(size_t)lds_scaled;
        const unsigned long long gp =
            (unsigned long long)(size_t)pred + (unsigned long long)row0 * 112ull;
        const unsigned long long gs =
            (unsigned long long)(size_t)scaled + (unsigned long long)row0 * 24ull;

        // Stage pred tile: 7 coalesced async b128 transfers per thread (NT: stream-once).
#pragma unroll
        for (int i = 0; i < 7; ++i) {
            const unsigned b = (unsigned)(i * TPB + tid) * 16u;
            const unsigned dst = lp + b;
            const unsigned long long src = gp + b;
            asm volatile("global_load_async_to_lds_b128 %0, %1, off th:TH_LOAD_NT"
                         :: "v"(dst), "v"(src) : "memory");
        }
        // Stage scaled tile: 384 async b128 transfers (1.5 per thread).
        {
            const unsigned b = (unsigned)tid * 16u;
            const unsigned dst = ls + b;
            const unsigned long long src = gs + b;
            asm volatile("global_load_async_to_lds_b128 %0, %1, off th:TH_LOAD_NT"
                         :: "v"(dst), "v"(src) : "memory");
        }
        if (tid < TPB / 2) {
            const unsigned b = (unsigned)(TPB + tid) * 16u;
            const unsigned dst = ls + b;
            const unsigned long long src = gs + b;
            asm volatile("global_load_async_to_lds_b128 %0, %1, off th:TH_LOAD_NT"
                         :: "v"(dst), "v"(src) : "memory");
        }
        // Wait for this wave's async transfers, then make tile visible WG-wide.
        asm volatile("s_wait_asynccnt 0x0" ::: "memory");
        __syncthreads();
    }

    const long long row = row0 + tid;
    if (row >= (long long)B) return;

    // Pull this thread's row into registers.
    union { float4 v4[7]; float f[28]; } p;
    float sp[6];
    if (full_tile) {
        const float4* q = (const float4*)(const void*)&lds_pred[tid * 28]; // 112B stride: 16B aligned
#pragma unroll
        for (int i = 0; i < 7; ++i) p.v4[i] = q[i];
#pragma unroll
        for (int i = 0; i < 6; ++i) sp[i] = lds_scaled[tid * 6 + i];
    } else {
#pragma unroll
        for (int i = 0; i < 28; ++i) p.f[i] = pred[row * 28 + i];
#pragma unroll
        for (int i = 0; i < 6; ++i) sp[i] = scaled[row * 6 + i];
    }

    // dp = scaled*scale + mean - bestfit
    float dp[6];
#pragma unroll
    for (int i = 0; i < 6; ++i)
        dp[i] = fmaf(sp[i], pscale[i], pmean[i]) - p.f[1 + i];

    // mahalanobis over symmetric C from upper triangle:
    //  m = sum_{i<=j} c_k * dp_i * dp_j * (i==j ? 1 : 2)
    const float* c = &p.f[7];
    float m = 0.0f;
    int k = 0;
#pragma unroll
    for (int i = 0; i < 6; ++i) {
#pragma unroll
        for (int j = i; j < 6; ++j) {
            const float w = (i == j) ? 1.0f : 2.0f;
            m = fmaf(c[k] * w, dp[i] * dp[j], m);
            ++k;
        }
    }

    const float v = fmaf(0.5f, m, p.f[0]);
    const float r = (v - tmean[0]) / tscale[0];
    __builtin_nontemporal_store(r, &out[row]);  // write-once output: NT store
}

extern "C" void kernel_launch(void* const* d_in, const int* in_sizes, int n_in,
                              void* d_out, int out_size, void* d_ws, size_t ws_size,
                              hipStream_t stream) {
    const float* scaled = (const float*)d_in[0];
    const float* pred   = (const float*)d_in[1];
    const float* pmean  = (const float*)d_in[2];
    const float* pscale = (const float*)d_in[3];
    const float* tmean  = (const float*)d_in[4];
    const float* tscale = (const float*)d_in[5];
    float* out = (float*)d_out;

    const int B = in_sizes[0] / 6;
    const int blocks = (B + TPB - 1) / TPB;
    ScaledPredictionLayer_kernel<<<blocks, TPB, 0, stream>>>(
        scaled, pred, pmean, pscale, tmean, tscale, out, B);
}